// GAT_Net_Class_35880156791104
// MI455X (gfx1250) — compile-verified
//
#include <hip/hip_runtime.h>
#include <math.h>

// ---------------------------------------------------------------------------
// GAT (2-layer) for MI455X / gfx1250.
// Dense projections: V_WMMA_F32_16X16X4_F32 (exact fp32, matches reference).
// Sparse attention: 3 edge passes with f32 global atomics; all row gathers
// vectorized to float4 (global_load_b128).
// ---------------------------------------------------------------------------

typedef float v2f __attribute__((ext_vector_type(2)));
typedef float v8f __attribute__((ext_vector_type(8)));

#define F_IN   256
#define HID    64      // H1*D1
#define H1     8
#define D1     8
#define NCLS   16      // H2*D2, single head
#define NEG_SLOPE 0.2f

// ---------------------------------------------------------------- utilities
__device__ inline void atomicMaxF32(float* addr, float val) {
    unsigned int* ua = (unsigned int*)addr;
    unsigned int old = *ua;
    while (__uint_as_float(old) < val) {
        unsigned int assumed = old;
        old = atomicCAS(ua, assumed, __float_as_uint(val));
        if (old == assumed) break;
    }
}

__device__ inline float lrelu(float v) { return v > 0.0f ? v : NEG_SLOPE * v; }

__global__ __launch_bounds__(256) void fill_kernel(float* __restrict__ p, float v, int n) {
    int i = blockIdx.x * blockDim.x + threadIdx.x;
    if (i < n) p[i] = v;
}

// ------------------------------------------------------- GEMM1: xp1 = x @ W1
// One block = 16 nodes (rows). 4 waves, each owns a 16x16 column tile of the
// 64 output columns. K-loop over 256 in steps of 4 using f32 WMMA.
__global__ __launch_bounds__(128) void gemm1_kernel(const float* __restrict__ x,
                                                    const float* __restrict__ W1,
                                                    float* __restrict__ xp1) {
    __shared__ float xs[16][F_IN + 4];            // +4 pad: row stride 260 (16B aligned)
    const int tid = threadIdx.x;
    const int nodeBase = blockIdx.x * 16;

    // cooperative b128 fill: 16x256 floats = 1024 float4, 8 per thread
    for (int i = tid; i < (16 * F_IN) / 4; i += 128) {
        int e4 = i * 4;
        int r = e4 >> 8, c = e4 & (F_IN - 1);
        *(float4*)(&xs[r][c]) = *(const float4*)(x + (size_t)(nodeBase + r) * F_IN + c);
    }
    __syncthreads();

    const int wave = tid >> 5;
    const int lane = tid & 31;
    const int m    = lane & 15;                   // A row
    const int kb   = (lane >> 4) * 2;             // A/B K sub-offset
    const int n    = lane & 15;                   // B/D column
    const int c0   = wave * 16;

    v8f acc = {};
    for (int k = 0; k < F_IN; k += 4) {
        v2f a, b;
        a.x = xs[m][k + kb];
        a.y = xs[m][k + kb + 1];
        b.x = W1[(k + kb    ) * HID + c0 + n];
        b.y = W1[(k + kb + 1) * HID + c0 + n];
        acc = __builtin_amdgcn_wmma_f32_16x16x4_f32(false, a, false, b,
                                                    (short)0, acc, false, false);
    }
    const int rowHalf = (lane >> 4) * 8;          // C/D layout: lanes>=16 hold M=8..15
    for (int r = 0; r < 8; ++r)
        xp1[(size_t)(nodeBase + rowHalf + r) * HID + c0 + n] = acc[r];
}

// ------------------------------------------------------ GEMM2: xp2 = h @ W2
// One block = 64 nodes; 4 waves each own one 16-node tile (16 output cols).
__global__ __launch_bounds__(128) void gemm2_kernel(const float* __restrict__ h,
                                                    const float* __restrict__ W2,
                                                    float* __restrict__ xp2, int N) {
    __shared__ float hs[64][HID + 4];             // row stride 68 floats (16B aligned)
    const int tid  = threadIdx.x;
    const int base = blockIdx.x * 64;

    for (int i = tid; i < (64 * HID) / 4; i += 128) {
        int e4 = i * 4;
        int r = e4 >> 6, c = e4 & (HID - 1);
        int nd = base + r;
        float4 v = {0.0f, 0.0f, 0.0f, 0.0f};
        if (nd < N) v = *(const float4*)(h + (size_t)nd * HID + c);
        *(float4*)(&hs[r][c]) = v;
    }
    __syncthreads();

    const int wave = tid >> 5;
    const int lane = tid & 31;
    const int tileRow = wave * 16;
    if (base + tileRow < N) {                     // wave-uniform guard
        const int m  = lane & 15;
        const int kb = (lane >> 4) * 2;
        const int n  = lane & 15;
        v8f acc = {};
        for (int k = 0; k < HID; k += 4) {
            v2f a, b;
            a.x = hs[tileRow + m][k + kb];
            a.y = hs[tileRow + m][k + kb + 1];
            b.x = W2[(k + kb    ) * NCLS + n];
            b.y = W2[(k + kb + 1) * NCLS + n];
            acc = __builtin_amdgcn_wmma_f32_16x16x4_f32(false, a, false, b,
                                                        (short)0, acc, false, false);
        }
        const int rowHalf = (lane >> 4) * 8;
        for (int r = 0; r < 8; ++r) {
            int nd = base + tileRow + rowHalf + r;
            if (nd < N) xp2[(size_t)nd * NCLS + n] = acc[r];
        }
    }
}

// ------------------------------------------------- node-level alpha dots L1
__global__ __launch_bounds__(256) void alpha1_kernel(const float* __restrict__ xp1,
                                                     const float* __restrict__ a_src,
                                                     const float* __restrict__ a_dst,
                                                     float* __restrict__ as1,
                                                     float* __restrict__ ad1, int N) {
    int idx = blockIdx.x * blockDim.x + threadIdx.x;      // over N*H1
    if (idx >= N * H1) return;
    int nd = idx >> 3, hh = idx & 7;
    const float4* row = (const float4*)(xp1 + (size_t)nd * HID + hh * D1);
    float4 r0 = row[0], r1 = row[1];
    const float* as = a_src + hh * D1;
    const float* ad = a_dst + hh * D1;
    float s = r0.x*as[0] + r0.y*as[1] + r0.z*as[2] + r0.w*as[3]
            + r1.x*as[4] + r1.y*as[5] + r1.z*as[6] + r1.w*as[7];
    float d = r0.x*ad[0] + r0.y*ad[1] + r0.z*ad[2] + r0.w*ad[3]
            + r1.x*ad[4] + r1.y*ad[5] + r1.z*ad[6] + r1.w*ad[7];
    as1[idx] = s; ad1[idx] = d;
}

__global__ __launch_bounds__(256) void alpha2_kernel(const float* __restrict__ xp2,
                                                     const float* __restrict__ a_src,
                                                     const float* __restrict__ a_dst,
                                                     float* __restrict__ as2,
                                                     float* __restrict__ ad2, int N) {
    int nd = blockIdx.x * blockDim.x + threadIdx.x;
    if (nd >= N) return;
    const float4* row = (const float4*)(xp2 + (size_t)nd * NCLS);
    float s = 0.0f, d = 0.0f;
    #pragma unroll
    for (int q = 0; q < 4; ++q) {
        float4 v = row[q];
        s += v.x*a_src[q*4+0] + v.y*a_src[q*4+1] + v.z*a_src[q*4+2] + v.w*a_src[q*4+3];
        d += v.x*a_dst[q*4+0] + v.y*a_dst[q*4+1] + v.z*a_dst[q*4+2] + v.w*a_dst[q*4+3];
    }
    as2[nd] = s; ad2[nd] = d;
}

// ---------------------------------------------------------- edge passes L1
__global__ __launch_bounds__(256) void edge_max1(const int* __restrict__ src,
                                                 const int* __restrict__ dst,
                                                 const float* __restrict__ as1,
                                                 const float* __restrict__ ad1,
                                                 float* __restrict__ m1, int E) {
    int idx = blockIdx.x * blockDim.x + threadIdx.x;      // over E*H1
    if (idx >= E * H1) return;
    int e = idx >> 3, hh = idx & 7;
    float v = lrelu(as1[src[e] * H1 + hh] + ad1[dst[e] * H1 + hh]);
    atomicMaxF32(&m1[dst[e] * H1 + hh], v);
}

__global__ __launch_bounds__(256) void fix_max(float* __restrict__ m, int n) {
    int i = blockIdx.x * blockDim.x + threadIdx.x;
    if (i < n && m[i] == -__builtin_inff()) m[i] = 0.0f;  // nodes with no in-edges
}

__global__ __launch_bounds__(256) void edge_sum1(const int* __restrict__ src,
                                                 const int* __restrict__ dst,
                                                 const float* __restrict__ as1,
                                                 const float* __restrict__ ad1,
                                                 const float* __restrict__ m1,
                                                 float* __restrict__ dn1, int E) {
    int idx = blockIdx.x * blockDim.x + threadIdx.x;
    if (idx >= E * H1) return;
    int e = idx >> 3, hh = idx & 7;
    int di = dst[e] * H1 + hh;
    float ex = __expf(lrelu(as1[src[e] * H1 + hh] + ad1[di]) - m1[di]);
    atomicAdd(&dn1[di], ex);
}

__global__ __launch_bounds__(256) void edge_agg1(const int* __restrict__ src,
                                                 const int* __restrict__ dst,
                                                 const float* __restrict__ as1,
                                                 const float* __restrict__ ad1,
                                                 const float* __restrict__ m1,
                                                 const float* __restrict__ dn1,
                                                 const float* __restrict__ xp1,
                                                 float* __restrict__ h1, int E) {
    int idx = blockIdx.x * blockDim.x + threadIdx.x;
    if (idx >= E * H1) return;
    int e = idx >> 3, hh = idx & 7;
    int s = src[e], d = dst[e];
    int di = d * H1 + hh;
    float ex    = __expf(lrelu(as1[s * H1 + hh] + ad1[di]) - m1[di]);
    float alpha = ex / (dn1[di] + 1e-16f);
    const float4* xr = (const float4*)(xp1 + (size_t)s * HID + hh * D1);  // 32B aligned
    float4 v0 = xr[0], v1 = xr[1];
    float* orow = h1 + (size_t)d * HID + hh * D1;
    atomicAdd(&orow[0], alpha * v0.x);
    atomicAdd(&orow[1], alpha * v0.y);
    atomicAdd(&orow[2], alpha * v0.z);
    atomicAdd(&orow[3], alpha * v0.w);
    atomicAdd(&orow[4], alpha * v1.x);
    atomicAdd(&orow[5], alpha * v1.y);
    atomicAdd(&orow[6], alpha * v1.z);
    atomicAdd(&orow[7], alpha * v1.w);
}

// bias + ELU in place on h1 (float4 per thread)
__global__ __launch_bounds__(256) void bias_elu(float* __restrict__ h1,
                                                const float* __restrict__ b1, int N) {
    int i = blockIdx.x * blockDim.x + threadIdx.x;        // over N*HID/4
    if (i >= N * (HID / 4)) return;
    int cq = i & (HID / 4 - 1);
    float4 v = ((float4*)h1)[i];
    float4 b = ((const float4*)b1)[cq];
    v.x += b.x; v.y += b.y; v.z += b.z; v.w += b.w;
    v.x = v.x > 0.0f ? v.x : (__expf(v.x) - 1.0f);
    v.y = v.y > 0.0f ? v.y : (__expf(v.y) - 1.0f);
    v.z = v.z > 0.0f ? v.z : (__expf(v.z) - 1.0f);
    v.w = v.w > 0.0f ? v.w : (__expf(v.w) - 1.0f);
    ((float4*)h1)[i] = v;
}

// ---------------------------------------------------------- edge passes L2
__global__ __launch_bounds__(256) void edge_max2(const int* __restrict__ src,
                                                 const int* __restrict__ dst,
                                                 const float* __restrict__ as2,
                                                 const float* __restrict__ ad2,
                                                 float* __restrict__ m2, int E) {
    int e = blockIdx.x * blockDim.x + threadIdx.x;
    if (e >= E) return;
    float v = lrelu(as2[src[e]] + ad2[dst[e]]);
    atomicMaxF32(&m2[dst[e]], v);
}

__global__ __launch_bounds__(256) void edge_sum2(const int* __restrict__ src,
                                                 const int* __restrict__ dst,
                                                 const float* __restrict__ as2,
                                                 const float* __restrict__ ad2,
                                                 const float* __restrict__ m2,
                                                 float* __restrict__ dn2, int E) {
    int e = blockIdx.x * blockDim.x + threadIdx.x;
    if (e >= E) return;
    int d = dst[e];
    float ex = __expf(lrelu(as2[src[e]] + ad2[d]) - m2[d]);
    atomicAdd(&dn2[d], ex);
}

__global__ __launch_bounds__(256) void edge_agg2(const int* __restrict__ src,
                                                 const int* __restrict__ dst,
                                                 const float* __restrict__ as2,
                                                 const float* __restrict__ ad2,
                                                 const float* __restrict__ m2,
                                                 const float* __restrict__ dn2,
                                                 const float* __restrict__ xp2,
                                                 float* __restrict__ o2, int E) {
    int e = blockIdx.x * blockDim.x + threadIdx.x;
    if (e >= E) return;
    int s = src[e], d = dst[e];
    float ex    = __expf(lrelu(as2[s] + ad2[d]) - m2[d]);
    float alpha = ex / (dn2[d] + 1e-16f);
    const float4* xr = (const float4*)(xp2 + (size_t)s * NCLS);           // 64B aligned
    float* orow = o2 + (size_t)d * NCLS;
    #pragma unroll
    for (int q = 0; q < 4; ++q) {
        float4 v = xr[q];
        atomicAdd(&orow[q*4+0], alpha * v.x);
        atomicAdd(&orow[q*4+1], alpha * v.y);
        atomicAdd(&orow[q*4+2], alpha * v.z);
        atomicAdd(&orow[q*4+3], alpha * v.w);
    }
}

// --------------------------------------------------- bias + log_softmax out
__global__ __launch_bounds__(256) void final_kernel(const float* __restrict__ o2,
                                                    const float* __restrict__ b2,
                                                    float* __restrict__ out, int N) {
    int nd = blockIdx.x * blockDim.x + threadIdx.x;
    if (nd >= N) return;
    const float4* row = (const float4*)(o2 + (size_t)nd * NCLS);
    const float4* bb  = (const float4*)b2;
    float v[NCLS];
    float mx = -__builtin_inff();
    #pragma unroll
    for (int q = 0; q < 4; ++q) {
        float4 r = row[q], b = bb[q];
        v[q*4+0] = r.x + b.x; v[q*4+1] = r.y + b.y;
        v[q*4+2] = r.z + b.z; v[q*4+3] = r.w + b.w;
    }
    #pragma unroll
    for (int k = 0; k < NCLS; ++k) mx = fmaxf(mx, v[k]);
    float sum = 0.0f;
    #pragma unroll
    for (int k = 0; k < NCLS; ++k) sum += __expf(v[k] - mx);
    float lse = __logf(sum);
    float4* o4 = (float4*)(out + (size_t)nd * NCLS);
    #pragma unroll
    for (int q = 0; q < 4; ++q) {
        float4 r;
        r.x = v[q*4+0] - mx - lse; r.y = v[q*4+1] - mx - lse;
        r.z = v[q*4+2] - mx - lse; r.w = v[q*4+3] - mx - lse;
        o4[q] = r;
    }
}

// ---------------------------------------------------------------------------
extern "C" void kernel_launch(void* const* d_in, const int* in_sizes, int n_in,
                              void* d_out, int out_size, void* d_ws, size_t ws_size,
                              hipStream_t stream) {
    const float* x   = (const float*)d_in[0];
    const int*   ei  = (const int*)  d_in[1];
    const float* W1  = (const float*)d_in[2];
    const float* asr1= (const float*)d_in[3];
    const float* ads1= (const float*)d_in[4];
    const float* b1  = (const float*)d_in[5];
    const float* W2  = (const float*)d_in[6];
    const float* asr2= (const float*)d_in[7];
    const float* ads2= (const float*)d_in[8];
    const float* b2  = (const float*)d_in[9];
    float* out = (float*)d_out;

    const int N = in_sizes[0] / F_IN;     // 100000
    const int E = in_sizes[1] / 2;        // 1600000
    const int* src = ei;
    const int* dst = ei + E;

    // workspace layout (floats); all sub-buffer offsets are multiples of N
    // floats (400000 B) -> 16B aligned for float4 access.
    float* ws  = (float*)d_ws;
    float* xp1 = ws;                      // N*64
    float* as1 = xp1 + (size_t)N * HID;   // N*8
    float* ad1 = as1 + (size_t)N * H1;
    float* m1  = ad1 + (size_t)N * H1;
    float* dn1 = m1  + (size_t)N * H1;
    float* h1  = dn1 + (size_t)N * H1;    // N*64 (accum -> ELU output)
    float* xp2 = h1  + (size_t)N * HID;   // N*16
    float* as2 = xp2 + (size_t)N * NCLS;  // N
    float* ad2 = as2 + (size_t)N;
    float* m2  = ad2 + (size_t)N;
    float* dn2 = m2  + (size_t)N;
    float* o2  = dn2 + (size_t)N;         // N*16

    const float NINF = -__builtin_inff();
    #define GRID(n) dim3(((n) + 255) / 256)

    // init accumulators (every call: graph-replay safe)
    fill_kernel<<<GRID(N * H1),  256, 0, stream>>>(m1,  NINF, N * H1);
    fill_kernel<<<GRID(N * H1),  256, 0, stream>>>(dn1, 0.0f, N * H1);
    fill_kernel<<<GRID(N * HID), 256, 0, stream>>>(h1,  0.0f, N * HID);
    fill_kernel<<<GRID(N),       256, 0, stream>>>(m2,  NINF, N);
    fill_kernel<<<GRID(N),       256, 0, stream>>>(dn2, 0.0f, N);
    fill_kernel<<<GRID(N * NCLS),256, 0, stream>>>(o2,  0.0f, N * NCLS);

    // ---- layer 1 ----
    gemm1_kernel<<<dim3(N / 16), 128, 0, stream>>>(x, W1, xp1);
    alpha1_kernel<<<GRID(N * H1), 256, 0, stream>>>(xp1, asr1, ads1, as1, ad1, N);
    edge_max1<<<GRID(E * H1), 256, 0, stream>>>(src, dst, as1, ad1, m1, E);
    fix_max<<<GRID(N * H1), 256, 0, stream>>>(m1, N * H1);
    edge_sum1<<<GRID(E * H1), 256, 0, stream>>>(src, dst, as1, ad1, m1, dn1, E);
    edge_agg1<<<GRID(E * H1), 256, 0, stream>>>(src, dst, as1, ad1, m1, dn1, xp1, h1, E);
    bias_elu<<<GRID(N * (HID / 4)), 256, 0, stream>>>(h1, b1, N);

    // ---- layer 2 ----
    gemm2_kernel<<<dim3((N + 63) / 64), 128, 0, stream>>>(h1, W2, xp2, N);
    alpha2_kernel<<<GRID(N), 256, 0, stream>>>(xp2, asr2, ads2, as2, ad2, N);
    edge_max2<<<GRID(E), 256, 0, stream>>>(src, dst, as2, ad2, m2, E);
    fix_max<<<GRID(N), 256, 0, stream>>>(m2, N);
    edge_sum2<<<GRID(E), 256, 0, stream>>>(src, dst, as2, ad2, m2, dn2, E);
    edge_agg2<<<GRID(E), 256, 0, stream>>>(src, dst, as2, ad2, m2, dn2, xp2, o2, E);

    // ---- output ----
    final_kernel<<<GRID(N), 256, 0, stream>>>(o2, b2, out, N);
    #undef GRID
}